// TemporalPatternExtractor_5248450035812
// MI455X (gfx1250) — compile-verified
//
#include <hip/hip_runtime.h>
#include <math.h>

// ---------------------------------------------------------------------------
// CDNA5 (gfx1250) implementation.
//  - f32 WMMA 16x16x4 for DFT and all attention matmuls
//  - Tensor Data Mover (tensor_load_to_lds, 6-arg clang-23 form) +
//    s_wait_tensorcnt for the DFT input tile
//  - wave32 layouts per CDNA5 ISA:
//      A 16x4 f32: lane l<16 -> (M=l, K=0/1), lane>=16 -> (M=l-16, K=2/3)
//      B  4x16 f32 (assumed dual): lane l -> (K = (l>>4)*2 + j, N = l&15)
//      C/D 16x16: vgpr r, lane l -> (M = r + (l>>4)*8, N = l&15)
// ---------------------------------------------------------------------------

typedef float v2f  __attribute__((ext_vector_type(2)));
typedef float v8f  __attribute__((ext_vector_type(8)));
typedef unsigned int u32x4 __attribute__((ext_vector_type(4)));
typedef int          i32x4 __attribute__((ext_vector_type(4)));
typedef int          i32x8 __attribute__((ext_vector_type(8)));

#define WMMA4(a, b, c) \
  __builtin_amdgcn_wmma_f32_16x16x4_f32(false, (a), false, (b), (short)0, (c), false, false)

static __device__ __forceinline__ v8f vzero8() {
  v8f z = {0.f, 0.f, 0.f, 0.f, 0.f, 0.f, 0.f, 0.f};
  return z;
}

// complex WMMA step: (cre,cim) += A(2 k-slices complex) * B(2 k-slices complex)
static __device__ __forceinline__ void cwmma(float2 a0, float2 a1,
                                             float2 b0, float2 b1,
                                             v8f& cre, v8f& cim) {
  v2f ar;  ar[0]  =  a0.x; ar[1]  =  a1.x;
  v2f ai;  ai[0]  =  a0.y; ai[1]  =  a1.y;
  v2f nai; nai[0] = -a0.y; nai[1] = -a1.y;
  v2f br;  br[0]  =  b0.x; br[1]  =  b1.x;
  v2f bi;  bi[0]  =  b0.y; bi[1]  =  b1.y;
  cre = WMMA4(ar,  br, cre);   // + Ar*Br
  cre = WMMA4(nai, bi, cre);   // - Ai*Bi   (f32 WMMA has no A/B neg -> negate in VALU)
  cim = WMMA4(ar,  bi, cim);   // + Ar*Bi
  cim = WMMA4(ai,  br, cim);   // + Ai*Br
}

// store C/D-layout accumulators into a row-major 16x16 complex LDS matrix
static __device__ __forceinline__ void store_c(float2* D, v8f cre, v8f cim, int lane) {
  int lr = lane & 15, lh = lane >> 4;
#pragma unroll
  for (int r = 0; r < 8; ++r)
    D[(r + lh * 8) * 16 + lr] = make_float2(cre[r], cim[r]);
}

// complex LayerNorm over C=16 (ddof=1), rows 0..14; zero row 15 (WMMA M-padding)
static __device__ __forceinline__ void ln_row(float2* M, const float2* w,
                                              const float2* b, int lane) {
  if (lane < 15) {
    int s = lane;
    float mr = 0.f, mi = 0.f;
    for (int c = 0; c < 16; ++c) { mr += M[s * 16 + c].x; mi += M[s * 16 + c].y; }
    mr *= (1.f / 16.f); mi *= (1.f / 16.f);
    float var = 0.f;
    for (int c = 0; c < 16; ++c) {
      float dr = M[s * 16 + c].x - mr, di = M[s * 16 + c].y - mi;
      var += dr * dr + di * di;
    }
    var *= (1.f / 15.f);
    float inv = rsqrtf(var);
    for (int c = 0; c < 16; ++c) {
      float zr = (M[s * 16 + c].x - mr) * inv;
      float zi = (M[s * 16 + c].y - mi) * inv;
      float2 wv_ = w[c], bv_ = b[c];
      M[s * 16 + c] = make_float2(wv_.x * zr - wv_.y * zi + bv_.x,
                                  wv_.x * zi + wv_.y * zr + bv_.y);
    }
  } else if (lane == 15) {
    for (int c = 0; c < 16; ++c) M[15 * 16 + c] = make_float2(0.f, 0.f);
  }
}

// ---------------------------------------------------------------------------
// Kernel 1: rfft(x)[...,:64] as real matmul with f32 WMMA.
// grid = 64 planes * 4 row-blocks; block = 256 threads (8 waves x 16 rows).
// 128x128 f32 tile (contiguous 64KB) DMA'd to LDS by the Tensor Data Mover.
// ---------------------------------------------------------------------------
__global__ __launch_bounds__(256) void k_dft(const float* __restrict__ x,
                                             float2* __restrict__ xf) {
  extern __shared__ char smem_raw[];
  float* xt = (float*)smem_raw;                 // [128][128]
  int tid = threadIdx.x;
  int p   = blockIdx.x >> 2;                    // plane = b*8+pw
  int n0  = (blockIdx.x & 3) * 128;

  if (tid < 32) {                               // wave 0 issues the TDM op
    unsigned lds_addr = (unsigned)(size_t)xt;   // generic->LDS byte offset (low 32b)
    unsigned long long ga =
        (unsigned long long)(size_t)(x + ((size_t)p * 512 + n0) * 128);
    u32x4 g0;
    g0[0] = 1u;                                              // count=1
    g0[1] = lds_addr;                                        // lds_addr [63:32]
    g0[2] = (unsigned)(ga & 0xFFFFFFFFu);                    // global_addr lo
    g0[3] = (unsigned)((ga >> 32) & 0x1FFFFFFu) | (2u << 30);// addr hi + type=2
    i32x8 g1;
    g1[0] = (int)(2u << 16);          // data_size=4B, no multicast/pad/iterate
    g1[1] = (int)(16384u << 16);      // tensor_dim0 = 16384 elements (low16)
    g1[2] = (int)(1u << 16);          // tensor_dim0 hi=0, tensor_dim1 = 1
    g1[3] = (int)(16384u << 16);      // tile_dim0 = 16384 (1-D 64KB tile)
    g1[4] = 0;                        // tile_dim1 = tile_dim2 = 0 (unused)
    g1[5] = 16384;                    // tensor_dim0_stride low32
    g1[6] = 0;
    g1[7] = 0;
    i32x4 gz; gz[0] = 0; gz[1] = 0; gz[2] = 0; gz[3] = 0;
    i32x8 gz8; gz8[0] = 0; gz8[1] = 0; gz8[2] = 0; gz8[3] = 0;
    gz8[4] = 0; gz8[5] = 0; gz8[6] = 0; gz8[7] = 0;
    __builtin_amdgcn_tensor_load_to_lds(g0, g1, gz, gz, gz8, 0);
    __builtin_amdgcn_s_wait_tensorcnt(0);
  }
  __syncthreads();

  int lane = tid & 31, wvi = tid >> 5;
  int lr = lane & 15, lh = lane >> 4;
  const float PI64 = 0.04908738521234052f;      // pi/64
  for (int ft = 0; ft < 4; ++ft) {
    int f = ft * 16 + lr;                       // B lane column (frequency)
    v8f cre = vzero8(), cim = vzero8();
    for (int kb = 0; kb < 128; kb += 4) {
      int k0 = kb + lh * 2;
      v2f a;
      a[0] = xt[(wvi * 16 + lr) * 128 + k0];
      a[1] = xt[(wvi * 16 + lr) * 128 + k0 + 1];
      float s0, c0, s1, c1v;
      __sincosf((float)((k0 * f) & 127) * PI64, &s0, &c0);
      __sincosf((float)(((k0 + 1) * f) & 127) * PI64, &s1, &c1v);
      v2f br; br[0] = c0;  br[1] = c1v;         // cos(2*pi*t*f/128)
      v2f bi; bi[0] = -s0; bi[1] = -s1;         // -sin(...)
      cre = WMMA4(a, br, cre);
      cim = WMMA4(a, bi, cim);
    }
#pragma unroll
    for (int r = 0; r < 8; ++r) {
      int row = n0 + wvi * 16 + r + lh * 8;
      xf[((size_t)p * 512 + row) * 64 + ft * 16 + lr] = make_float2(cre[r], cim[r]);
    }
  }
}

// ---------------------------------------------------------------------------
// Kernel 2a: fold ef1 into conv1:  A1[o][pw][to][k] = sum_ne conv1_w * ef1_w
// ---------------------------------------------------------------------------
__global__ __launch_bounds__(256) void k_mkA1(const float2* __restrict__ conv1w,
                                              const float2* __restrict__ ef1w,
                                              float2* __restrict__ A1) {
  for (int e = threadIdx.x; e < 16 * 8 * 31 * 3; e += 256) {
    int k = e % 3, to = (e / 3) % 31, pw = (e / 93) % 8, o = e / 744;
    float ar = 0.f, ai = 0.f;
    for (int ne = 0; ne < 16; ++ne) {
      float2 w  = conv1w[(o * 128 + ne * 8 + pw) * 3 + k];
      float2 ef = ef1w[(ne * 8 + pw) * 64 + 2 * to + k];
      ar += w.x * ef.x - w.y * ef.y;
      ai += w.x * ef.y + w.y * ef.x;
    }
    A1[e] = make_float2(ar, ai);   // layout ((o*8+pw)*31+to)*3+k == e
  }
}

// ---------------------------------------------------------------------------
// Kernel 2b: c1[b,o,n,to] = sum_{pw,k} A1 * xf[b*8+pw, n, 2to+k]
// grid = 8 b * 32 n-tiles; block (o=16 x dn=16)
// ---------------------------------------------------------------------------
__global__ __launch_bounds__(256) void k_conv1(const float2* __restrict__ xf,
                                               const float2* __restrict__ A1,
                                               float2* __restrict__ c1) {
  extern __shared__ char smem_raw[];
  float2* xs = (float2*)smem_raw;               // [8][16][64]
  int tid = threadIdx.x;
  int b = blockIdx.x >> 5;
  int n0 = (blockIdx.x & 31) * 16;
  __builtin_prefetch(A1 + (tid * 3) % 11904, 0, 1);
  for (int i = tid; i < 8 * 16 * 64; i += 256) {
    int pw = i >> 10, dn = (i >> 6) & 15, t = i & 63;
    xs[i] = xf[((size_t)(b * 8 + pw) * 512 + n0 + dn) * 64 + t];
  }
  __syncthreads();
  int o = tid >> 4, dn = tid & 15;
  for (int to = 0; to < 31; ++to) {
    float ar = 0.f, ai = 0.f;
    for (int pw = 0; pw < 8; ++pw)
#pragma unroll
      for (int k = 0; k < 3; ++k) {
        float2 w = A1[((o * 8 + pw) * 31 + to) * 3 + k];
        float2 v = xs[(pw * 16 + dn) * 64 + 2 * to + k];
        ar += w.x * v.x - w.y * v.y;
        ai += w.x * v.y + w.y * v.x;
      }
    c1[((size_t)(b * 16 + o) * 512 + n0 + dn) * 31 + to] = make_float2(ar, ai);
  }
}

// ---------------------------------------------------------------------------
// Kernel 3: freq_emb[b,o,n,t2] = sum_{i,k} conv2_w[o,i,k]*ef2_w[i,2t2+k]*c1[b,i,n,2t2+k]
// Also emits xe[bp][s][c] for the attention stage.
// ---------------------------------------------------------------------------
__global__ __launch_bounds__(256) void k_conv2(const float2* __restrict__ c1,
                                               const float2* __restrict__ conv2w,
                                               const float2* __restrict__ ef2w,
                                               float2* __restrict__ fe_out,
                                               float2* __restrict__ xe) {
  extern __shared__ char smem_raw[];
  float2* cs = (float2*)smem_raw;               // [16][16][31]
  int tid = threadIdx.x;
  int b = blockIdx.x >> 5;
  int n0 = (blockIdx.x & 31) * 16;
  for (int i = tid; i < 16 * 16 * 31; i += 256) {
    int ic = i / 496, rem = i % 496, dn = rem / 31, t1 = rem % 31;
    cs[(ic * 16 + dn) * 31 + t1] =
        c1[((size_t)(b * 16 + ic) * 512 + n0 + dn) * 31 + t1];
  }
  __syncthreads();
  int o = tid >> 4, dn = tid & 15, n = n0 + dn;
  for (int t2 = 0; t2 < 15; ++t2) {
    float ar = 0.f, ai = 0.f;
    for (int ic = 0; ic < 16; ++ic)
#pragma unroll
      for (int k = 0; k < 3; ++k) {
        float2 cw = conv2w[(o * 16 + ic) * 3 + k];
        float2 ew = ef2w[ic * 31 + 2 * t2 + k];
        float wr = cw.x * ew.x - cw.y * ew.y;
        float wi = cw.x * ew.y + cw.y * ew.x;
        float2 v = cs[(ic * 16 + dn) * 31 + 2 * t2 + k];
        ar += wr * v.x - wi * v.y;
        ai += wr * v.y + wi * v.x;
      }
    float2 r = make_float2(ar, ai);
    fe_out[((size_t)(b * 16 + o) * 512 + n) * 15 + t2] = r;     // freq_emb output
    xe[(((size_t)(b * 512 + n)) * 15 + t2) * 16 + o] = r;       // [bp][s][c]
  }
}

// ---------------------------------------------------------------------------
// Kernel 4: per batch-row complex MHA + cLN + FFN + cLN.  1 wave = 1 bp.
// All matmuls via f32 WMMA; transposes bounced through per-wave LDS scratch
// (DS ops are in-order within a wave -> no barriers needed for wave-private LDS).
// ---------------------------------------------------------------------------
__global__ __launch_bounds__(128) void k_attn(
    const float2* __restrict__ xe,
    const float2* __restrict__ wq, const float2* __restrict__ bq,
    const float2* __restrict__ wk, const float2* __restrict__ bk,
    const float2* __restrict__ wv, const float2* __restrict__ bv,
    const float2* __restrict__ wcat, const float2* __restrict__ bcat,
    const float2* __restrict__ ln1w, const float2* __restrict__ ln1b,
    const float2* __restrict__ ffnw, const float2* __restrict__ ffnb,
    const float2* __restrict__ ln2w, const float2* __restrict__ ln2b,
    float2* __restrict__ out, float2* __restrict__ score_out) {
  extern __shared__ char smem_raw[];
  float2* S  = (float2*)smem_raw;
  float2* WQ = S;                 // [64][16]
  float2* WK = S + 1024;          // [64][16]
  float2* WV = S + 2048;          // [64][16]
  float2* WC = S + 3072;          // [16][64] (wcat as given)
  float2* FT = S + 4096;          // [16][16]  FT[k][n] = ffn_w[n][k]
  int tid = threadIdx.x;
  for (int i = tid; i < 1024; i += 128) {
    WQ[i] = wq[i]; WK[i] = wk[i]; WV[i] = wv[i]; WC[i] = wcat[i];
  }
  for (int i = tid; i < 256; i += 128) {
    int kk = i >> 4, nn_ = i & 15;
    FT[i] = ffnw[nn_ * 16 + kk];
  }
  __syncthreads();

  int wvi = tid >> 5, lane = tid & 31;
  int lr = lane & 15, lh = lane >> 4;
  int bp = blockIdx.x * 4 + wvi;

  float2* WB = S + 4352 + wvi * 1664;   // per-wave scratch
  float2* XE = WB;                      // [s][c]
  float2* Qb = WB + 256;                // [s][e_h]
  float2* Kb = WB + 512;                // conj(k)^T: [c][t]
  float2* Vb = WB + 768;                // conj(v):   [t][c]
  float2* Ab = WB + 1024;               // av / ffn-out scratch
  float2* Hb = WB + 1280;               // residual / LN buffer
  float*  SC = (float*)(WB + 1536);     // score [s][t] (real)

  for (int i = lane; i < 256; i += 32) {
    int s = i >> 4, c = i & 15;
    XE[i] = (s < 15) ? xe[((size_t)bp * 15 + s) * 16 + c] : make_float2(0.f, 0.f);
  }

  v8f atr = vzero8(), ati = vzero8();   // attn_out accumulator across heads

  for (int h = 0; h < 4; ++h) {
    // ---- q_h = XE @ Wq_h^T + bq_h  (C-layout [s][e])
    v8f qre = vzero8(), qim = vzero8();
    for (int kb = 0; kb < 16; kb += 4) {
      int k0 = kb + lh * 2;
      cwmma(XE[lr * 16 + k0], XE[lr * 16 + k0 + 1],
            WQ[(h * 16 + lr) * 16 + k0], WQ[(h * 16 + lr) * 16 + k0 + 1],
            qre, qim);
    }
    {
      float2 bv_ = bq[h * 16 + lr];
#pragma unroll
      for (int r = 0; r < 8; ++r) { qre[r] += bv_.x; qim[r] += bv_.y; }
    }
    store_c(Qb, qre, qim, lane);

    // ---- conj(k_h)^T = conj(Wk_h @ XE^T + bk_h)   [c][t]
    // XE's A-layout registers double as XE^T's B-layout registers.
    v8f kre = vzero8(), kim = vzero8();
    for (int kb = 0; kb < 16; kb += 4) {
      int k0 = kb + lh * 2;
      cwmma(WK[(h * 16 + lr) * 16 + k0], WK[(h * 16 + lr) * 16 + k0 + 1],
            XE[lr * 16 + k0], XE[lr * 16 + k0 + 1], kre, kim);
    }
#pragma unroll
    for (int r = 0; r < 8; ++r) {
      float2 bv_ = bk[h * 16 + r + lh * 8];   // row bias (row = channel)
      kre[r] += bv_.x;
      kim[r] = -(kim[r] + bv_.y);             // conjugate
    }
    store_c(Kb, kre, kim, lane);

    // ---- conj(v_h)  [t][c]
    v8f vre = vzero8(), vim = vzero8();
    for (int kb = 0; kb < 16; kb += 4) {
      int k0 = kb + lh * 2;
      cwmma(XE[lr * 16 + k0], XE[lr * 16 + k0 + 1],
            WV[(h * 16 + lr) * 16 + k0], WV[(h * 16 + lr) * 16 + k0 + 1],
            vre, vim);
    }
    {
      float2 bv_ = bv[h * 16 + lr];
#pragma unroll
      for (int r = 0; r < 8; ++r) { vre[r] += bv_.x; vim[r] = -(vim[r] + bv_.y); }
    }
    store_c(Vb, vre, vim, lane);

    // ---- sim = q_h @ conj(k_h)^T
    v8f sre = vzero8(), sim2 = vzero8();
    for (int kb = 0; kb < 16; kb += 4) {
      int k0 = kb + lh * 2;
      cwmma(Qb[lr * 16 + k0], Qb[lr * 16 + k0 + 1],
            Kb[k0 * 16 + lr], Kb[(k0 + 1) * 16 + lr], sre, sim2);
    }
    const float isq = 0.1767766952966369f;    // 1/sqrt(2*ch)
#pragma unroll
    for (int r = 0; r < 8; ++r)
      SC[(r + lh * 8) * 16 + lr] = sqrtf(sre[r] * sre[r] + sim2[r] * sim2[r]) * isq;

    // ---- softmax over t (rows handled by lanes), emit score output
    if (lane < 15) {
      int s = lane;
      float mx = -1e30f;
      for (int t = 0; t < 15; ++t) mx = fmaxf(mx, SC[s * 16 + t]);
      float sum = 0.f;
      for (int t = 0; t < 15; ++t) {
        float e = __expf(SC[s * 16 + t] - mx);
        SC[s * 16 + t] = e;
        sum += e;
      }
      float inv = 1.f / sum;
      for (int t = 0; t < 15; ++t) {
        float v_ = SC[s * 16 + t] * inv;
        SC[s * 16 + t] = v_;
        score_out[(((size_t)bp * 4 + h) * 15 + s) * 15 + t] = make_float2(v_, 0.f);
      }
      SC[s * 16 + 15] = 0.f;                  // K padding column
    } else if (lane == 15) {
      for (int t = 0; t < 16; ++t) SC[15 * 16 + t] = 0.f;  // M padding row
    }

    // ---- av = score @ conj(v)   (real A x complex B)
    v8f avr = vzero8(), avi = vzero8();
    for (int kb = 0; kb < 16; kb += 4) {
      int k0 = kb + lh * 2;
      v2f a; a[0] = SC[lr * 16 + k0]; a[1] = SC[lr * 16 + k0 + 1];
      float2 b0 = Vb[k0 * 16 + lr], b1 = Vb[(k0 + 1) * 16 + lr];
      v2f br; br[0] = b0.x; br[1] = b1.x;
      v2f bi; bi[0] = b0.y; bi[1] = b1.y;
      avr = WMMA4(a, br, avr);
      avi = WMMA4(a, bi, avi);
    }
    store_c(Ab, avr, avi, lane);

    // ---- attn += av_h @ wcat_h^T
    for (int kb = 0; kb < 16; kb += 4) {
      int k0 = kb + lh * 2;
      cwmma(Ab[lr * 16 + k0], Ab[lr * 16 + k0 + 1],
            WC[lr * 64 + h * 16 + k0], WC[lr * 64 + h * 16 + k0 + 1],
            atr, ati);
    }
  }

  // + bcat + residual xe, then cLN1
  {
    float2 bv_ = bcat[lr];
#pragma unroll
    for (int r = 0; r < 8; ++r) {
      float2 xv = XE[(r + lh * 8) * 16 + lr];
      atr[r] += bv_.x + xv.x;
      ati[r] += bv_.y + xv.y;
    }
  }
  store_c(Hb, atr, ati, lane);
  ln_row(Hb, ln1w, ln1b, lane);

  // ffn: h @ ffn_w^T + ffn_b + h, then cLN2
  v8f fre = vzero8(), fim = vzero8();
  for (int kb = 0; kb < 16; kb += 4) {
    int k0 = kb + lh * 2;
    cwmma(Hb[lr * 16 + k0], Hb[lr * 16 + k0 + 1],
          FT[k0 * 16 + lr], FT[(k0 + 1) * 16 + lr], fre, fim);
  }
  {
    float2 bv_ = ffnb[lr];
#pragma unroll
    for (int r = 0; r < 8; ++r) {
      float2 hv = Hb[(r + lh * 8) * 16 + lr];
      fre[r] += bv_.x + hv.x;
      fim[r] += bv_.y + hv.y;
    }
  }
  store_c(Ab, fre, fim, lane);
  ln_row(Ab, ln2w, ln2b, lane);

  if (lane < 15) {
    int s = lane;
    for (int c = 0; c < 16; ++c)
      out[(size_t)bp * 240 + s * 16 + c] = Ab[s * 16 + c];
  }
}

// ---------------------------------------------------------------------------
extern "C" void kernel_launch(void* const* d_in, const int* in_sizes, int n_in,
                              void* d_out, int out_size, void* d_ws, size_t ws_size,
                              hipStream_t stream) {
  (void)in_sizes; (void)n_in; (void)out_size; (void)ws_size;
  const float*  x      = (const float*)d_in[0];
  const float2* ef1_w  = (const float2*)d_in[1];
  const float2* conv1w = (const float2*)d_in[2];
  const float2* ef2_w  = (const float2*)d_in[3];
  const float2* conv2w = (const float2*)d_in[4];
  const float2* wq   = (const float2*)d_in[5];
  const float2* bq   = (const float2*)d_in[6];
  const float2* wk   = (const float2*)d_in[7];
  const float2* bk   = (const float2*)d_in[8];
  const float2* wv   = (const float2*)d_in[9];
  const float2* bv   = (const float2*)d_in[10];
  const float2* wcat = (const float2*)d_in[11];
  const float2* bcat = (const float2*)d_in[12];
  const float2* ln1w = (const float2*)d_in[13];
  const float2* ln1b = (const float2*)d_in[14];
  const float2* ffnw = (const float2*)d_in[15];
  const float2* ffnb = (const float2*)d_in[16];
  const float2* ln2w = (const float2*)d_in[17];
  const float2* ln2b = (const float2*)d_in[18];

  float2* ws = (float2*)d_ws;
  float2* xf = ws;                 // 64*512*64          = 2,097,152 cplx
  float2* c1 = ws + 2097152;       // 8*16*512*31        = 2,031,616 cplx
  float2* A1 = ws + 4128768;       // 16*8*31*3          =    11,904 cplx
  float2* xe = ws + 4140672;       // 4096*15*16         =   983,040 cplx

  float2* outp = (float2*)d_out;   // out:      8*512*240   = 983,040 cplx
  float2* fe   = outp + 983040;    // freq_emb: 8*16*512*15 = 983,040 cplx
  float2* sc   = outp + 1966080;   // score:    4096*4*15*15= 3,686,400 cplx

  hipLaunchKernelGGL(k_dft,   dim3(256),  dim3(256), 65536, stream, x, xf);
  hipLaunchKernelGGL(k_mkA1,  dim3(1),    dim3(256), 0,     stream, conv1w, ef1_w, A1);
  hipLaunchKernelGGL(k_conv1, dim3(256),  dim3(256), 65536, stream, xf, A1, c1);
  hipLaunchKernelGGL(k_conv2, dim3(256),  dim3(256), 63488, stream, c1, conv2w, ef2_w, fe, xe);
  hipLaunchKernelGGL(k_attn,  dim3(1024), dim3(128), 88064, stream, xe,
                     wq, bq, wk, bk, wv, bv, wcat, bcat,
                     ln1w, ln1b, ffnw, ffnb, ln2w, ln2b, outp, sc);
}